// GATEncoder_32057635897400
// MI455X (gfx1250) — compile-verified
//
#include <hip/hip_runtime.h>
#include <hip/hip_bf16.h>

// ---------------------------------------------------------------------------
// GAT encoder for MI455X (gfx1250, wave32).
// GEMMs run on the matrix pipe via V_WMMA_F32_16X16X4_F32, register-blocked
// 16x64 per wave. The 64-wide B strip is staged through LDS once per 8-wave
// block in a pair-interleaved layout so each WMMA B-fragment is a single
// conflict-free ds_load_b64; A fragments for the whole K-chunk are hoisted
// into registers as one load clause. Next B chunk prefetched with
// global_prefetch_b8. fp32 end-to-end keeps reference precision; the edge
// gather/scatter (~2 GB of L2/HBM traffic) remains the roofline bound.
// ---------------------------------------------------------------------------

#define N_NODES 20000
#define IN_DIM  128
#define C1      256
#define H1      3
#define HC1     768        // H1*C1
#define C2      384
#define E_BASE  320000
#define E_TOT   340000     // + self loops
#define NEG_SLOPE 0.2f

#define KB      32         // K-chunk staged in LDS (16 K-pairs)
#define PRSTRIDE 160       // floats per LDS pair-row (64 pairs = 128 + 32 pad)
                           // pad of 16 pairs -> p and p+1 on opposite bank halves

typedef __attribute__((ext_vector_type(2))) float v2f;
typedef __attribute__((ext_vector_type(8))) float v8f;

// ---------------------------------------------------------------------------
// fp32 WMMA GEMM: C[M,N] = A[M,K] * B[K,N].
// Block = 8 waves = 8 consecutive 16-row tiles x one 64-col strip (blockIdx.y).
// LDS holds B chunk as pairs: float index (p*PRSTRIDE + 2*c + sub) = B[2p+sub][c].
// VGPR layouts per CDNA5 ISA 7.12.2:
//   A 16x4 : lanes 0-15 hold M=lane {K=k,k+1}; lanes 16-31 hold {K=k+2,k+3}
//   B 4x16 : v0/v1 = rows K=k,k+1 (lanes 0-15) / K=k+2,k+3 (lanes 16-31)
//   C 16x16: VGPR r, lanes 0-15 -> M=r, lanes 16-31 -> M=r+8, N=lane&15
// N%64==0, K%KB==0 for all problem sizes -> no K/N tails; EXEC all ones.
// ---------------------------------------------------------------------------
__global__ void wmma_gemm_f32(const float* __restrict__ A,
                              const float* __restrict__ B,
                              float* __restrict__ C,
                              int M, int N, int K) {
    __shared__ float bs[(KB / 2) * PRSTRIDE];         // 16 pair-rows, 10 KB

    const int wave   = threadIdx.x >> 5;
    const int lane   = threadIdx.x & 31;
    const int tilesM = M >> 4;
    const int tn     = blockIdx.y;                    // 64-col strip
    const int tm     = blockIdx.x * 8 + wave;
    const int tmc    = tm < tilesM ? tm : tilesM - 1; // clamp; idle waves skip store

    const int l  = lane & 15;
    const int kh = (lane >> 4) << 1;                  // 0 (lanes 0-15) / 2 (16-31)
    const int ph = (lane >> 4);                       // pair offset for high lanes

    const float* __restrict__ arow   = A + (size_t)(tmc * 16 + l) * K + kh;
    const float* __restrict__ bstrip = B + (size_t)(tn * 64);

    v8f acc0 = {}, acc1 = {}, acc2 = {}, acc3 = {};

    for (int kc = 0; kc < K; kc += KB) {
        __syncthreads();
        // Cooperative stage, pair-interleaved: pos = p*128 + 2*c + sub.
        // Consecutive threads write consecutive LDS floats (conflict-free).
#pragma unroll
        for (int i = 0; i < (KB * 64) / 256; ++i) {
            const int pos = threadIdx.x + i * 256;
            const int p   = pos >> 7;                 // pair-row
            const int rem = pos & 127;                // 2*c + sub
            const int c   = rem >> 1;
            const int sub = rem & 1;
            bs[p * PRSTRIDE + rem] = bstrip[(size_t)(kc + 2 * p + sub) * N + c];
        }
        __syncthreads();

        // Prefetch next B chunk (64 x 128B cachelines) while computing.
        if (kc + KB < K && threadIdx.x < 64)
            __builtin_prefetch(&bstrip[(size_t)(kc + KB + (threadIdx.x >> 1)) * N +
                                       (threadIdx.x & 1) * 32], 0, 3);

        // Hoist all A fragments of this chunk into registers (one clause).
        v2f afr[KB / 4];
#pragma unroll
        for (int k8 = 0; k8 < KB / 4; ++k8) {
            afr[k8].x = arow[kc + 4 * k8];
            afr[k8].y = arow[kc + 4 * k8 + 1];
        }

#pragma unroll
        for (int k8 = 0; k8 < KB / 4; ++k8) {
            // pair index: lanes 0-15 -> 2*k8, lanes 16-31 -> 2*k8+1
            const float* __restrict__ bp = bs + (2 * k8 + ph) * PRSTRIDE + 2 * l;
            const v2f b0 = *(const v2f*)(bp + 0);
            const v2f b1 = *(const v2f*)(bp + 32);
            const v2f b2 = *(const v2f*)(bp + 64);
            const v2f b3 = *(const v2f*)(bp + 96);
            acc0 = __builtin_amdgcn_wmma_f32_16x16x4_f32(false, afr[k8], false, b0, (short)0, acc0, false, false);
            acc1 = __builtin_amdgcn_wmma_f32_16x16x4_f32(false, afr[k8], false, b1, (short)0, acc1, false, false);
            acc2 = __builtin_amdgcn_wmma_f32_16x16x4_f32(false, afr[k8], false, b2, (short)0, acc2, false, false);
            acc3 = __builtin_amdgcn_wmma_f32_16x16x4_f32(false, afr[k8], false, b3, (short)0, acc3, false, false);
        }
    }

    if (tm >= tilesM) return;                         // wave-uniform

    const int hi = (lane >> 4) << 3;                  // +8 row offset, high lanes
    float* __restrict__ crow = C + (size_t)(tm * 16) * N + tn * 64 + l;
#pragma unroll
    for (int r = 0; r < 8; ++r) {
        const size_t off = (size_t)(r + hi) * N;
        crow[off +  0] = acc0[r];
        crow[off + 16] = acc1[r];
        crow[off + 32] = acc2[r];
        crow[off + 48] = acc3[r];
    }
}

// ---------------------------------------------------------------------------
// Per-node attention logits: als[n,h] = sum_c h[n,h*C+c]*att_s[h,c]  (same att_d)
// ---------------------------------------------------------------------------
__global__ void logits_kernel(const float* __restrict__ h,
                              const float* __restrict__ att_s,
                              const float* __restrict__ att_d,
                              float* __restrict__ als,
                              float* __restrict__ ald,
                              int n_nodes, int H, int C) {
    const int i = blockIdx.x * blockDim.x + threadIdx.x;
    if (i >= n_nodes * H) return;
    const int n = i / H;
    const int hh = i % H;
    const float* __restrict__ row = h + (size_t)n * H * C + (size_t)hh * C;
    const float* __restrict__ vs  = att_s + (size_t)hh * C;
    const float* __restrict__ vd  = att_d + (size_t)hh * C;
    float ss = 0.f, sd = 0.f;
    for (int c = 0; c < C; ++c) {
        const float v = row[c];
        ss += v * vs[c];
        sd += v * vd[c];
    }
    als[i] = ss;
    ald[i] = sd;
}

__global__ void fill_neginf(float* __restrict__ p, int n) {
    const int i = blockIdx.x * blockDim.x + threadIdx.x;
    if (i < n) p[i] = -__builtin_inff();
}

__device__ __forceinline__ void atomicMaxF(float* addr, float val) {
    // Monotonic IEEE-754 bit trick: signed max for val>=0, unsigned min for <0.
    if (val >= 0.f) atomicMax((int*)addr, __float_as_int(val));
    else            atomicMin((unsigned int*)addr, __float_as_uint(val));
}

__device__ __forceinline__ void edge_endpoints(int e, const int* __restrict__ src,
                                               const int* __restrict__ dst,
                                               int& s, int& d) {
    if (e < E_BASE) { s = src[e]; d = dst[e]; }
    else            { s = e - E_BASE; d = s; }           // self loops
}

__device__ __forceinline__ float leaky(float x) {
    return x > 0.f ? x : NEG_SLOPE * x;
}

// Pass A: segment max of leaky_relu(als[src]+ald[dst]) grouped by dst.
__global__ void edge_max_kernel(const int* __restrict__ src, const int* __restrict__ dst,
                                const float* __restrict__ als, const float* __restrict__ ald,
                                float* __restrict__ emax, int H) {
    const int e = blockIdx.x * blockDim.x + threadIdx.x;
    if (e >= E_TOT) return;
    int s, d; edge_endpoints(e, src, dst, s, d);
    for (int hh = 0; hh < H; ++hh) {
        const float v = leaky(als[s * H + hh] + ald[d * H + hh]);
        atomicMaxF(&emax[d * H + hh], v);
    }
}

// Pass B: e_exp = exp(e - max); denom[dst] += e_exp.
__global__ void edge_exp_kernel(const int* __restrict__ src, const int* __restrict__ dst,
                                const float* __restrict__ als, const float* __restrict__ ald,
                                const float* __restrict__ emax,
                                float* __restrict__ eexp, float* __restrict__ denom, int H) {
    const int e = blockIdx.x * blockDim.x + threadIdx.x;
    if (e >= E_TOT) return;
    int s, d; edge_endpoints(e, src, dst, s, d);
    for (int hh = 0; hh < H; ++hh) {
        const float v  = leaky(als[s * H + hh] + ald[d * H + hh]);
        const float ex = __expf(v - emax[d * H + hh]);
        eexp[(size_t)e * H + hh] = ex;
        atomicAdd(&denom[d * H + hh], ex);
    }
}

// Pass C: agg[dst, h*C+c] += h[src, h*C+c] * (eexp / denom[dst,h]); block per edge.
// Head-outer loop keeps the inner loop pure FMA + global_atomic_add_f32.
__global__ void edge_scatter_kernel(const int* __restrict__ src, const int* __restrict__ dst,
                                    const float* __restrict__ hsrc,
                                    const float* __restrict__ eexp,
                                    const float* __restrict__ denom,
                                    float* __restrict__ agg, int H, int C) {
    const int e = blockIdx.x;
    int s, d; edge_endpoints(e, src, dst, s, d);
    __shared__ float alpha[8];
    if (threadIdx.x < H)
        alpha[threadIdx.x] = eexp[(size_t)e * H + threadIdx.x] / denom[d * H + threadIdx.x];
    __syncthreads();
    const float* __restrict__ row = hsrc + (size_t)s * H * C;
    float* __restrict__ out = agg + (size_t)d * H * C;
    for (int hh = 0; hh < H; ++hh) {
        const float a = alpha[hh];
        const int base = hh * C;
        for (int c = threadIdx.x; c < C; c += blockDim.x)
            atomicAdd(&out[base + c], row[base + c] * a);
    }
}

// out[row, col] = relu(out[row, col] + bias[col]); grid = (cols/256, rows)
__global__ void bias_relu_kernel(float* __restrict__ p, const float* __restrict__ bias,
                                 int stride) {
    const int col = blockIdx.x * blockDim.x + threadIdx.x;
    if (col >= stride) return;
    const size_t i = (size_t)blockIdx.y * stride + col;
    p[i] = fmaxf(p[i] + bias[col], 0.f);
}

// ---------------------------------------------------------------------------
extern "C" void kernel_launch(void* const* d_in, const int* in_sizes, int n_in,
                              void* d_out, int out_size, void* d_ws, size_t ws_size,
                              hipStream_t stream) {
    const float* x   = (const float*)d_in[0];
    const int*   ei  = (const int*)  d_in[1];
    const float* W1  = (const float*)d_in[2];
    const float* as1 = (const float*)d_in[3];
    const float* ad1 = (const float*)d_in[4];
    const float* b1  = (const float*)d_in[5];
    const float* W2  = (const float*)d_in[6];
    const float* as2 = (const float*)d_in[7];
    const float* ad2 = (const float*)d_in[8];
    const float* b2  = (const float*)d_in[9];
    float* out = (float*)d_out;

    const int* srcI = ei;
    const int* dstI = ei + E_BASE;

    // Workspace carve-up (~160 MB total)
    char* ws = (char*)d_ws;
    float* h1    = (float*)ws; ws += (size_t)N_NODES * HC1 * 4;  // 61.44 MB
    float* agg1  = (float*)ws; ws += (size_t)N_NODES * HC1 * 4;  // 61.44 MB
    float* h2    = (float*)ws; ws += (size_t)N_NODES * C2  * 4;  // 30.72 MB
    float* als_1 = (float*)ws; ws += (size_t)N_NODES * H1 * 4;
    float* ald_1 = (float*)ws; ws += (size_t)N_NODES * H1 * 4;
    float* emax1 = (float*)ws; ws += (size_t)N_NODES * H1 * 4;
    float* den1  = (float*)ws; ws += (size_t)N_NODES * H1 * 4;
    float* eexp1 = (float*)ws; ws += (size_t)E_TOT  * H1 * 4;    // 4.08 MB
    float* als_2 = (float*)ws; ws += (size_t)N_NODES * 4;
    float* ald_2 = (float*)ws; ws += (size_t)N_NODES * 4;
    float* emax2 = (float*)ws; ws += (size_t)N_NODES * 4;
    float* den2  = (float*)ws; ws += (size_t)N_NODES * 4;
    float* eexp2 = (float*)ws; ws += (size_t)E_TOT  * 4;         // 1.36 MB

    const int edgeBlocks = (E_TOT + 255) / 256;
    const int tilesM = N_NODES / 16;                  // 1250

    // ---------------- Layer 1 ----------------
    {   // h1 = x @ W1   (M=20000, N=768, K=128)
        dim3 grid((tilesM + 7) / 8, HC1 / 64);
        wmma_gemm_f32<<<grid, 256, 0, stream>>>(x, W1, h1, N_NODES, HC1, IN_DIM);
    }
    logits_kernel<<<(N_NODES * H1 + 255) / 256, 256, 0, stream>>>(h1, as1, ad1, als_1, ald_1,
                                                                  N_NODES, H1, C1);
    fill_neginf<<<(N_NODES * H1 + 255) / 256, 256, 0, stream>>>(emax1, N_NODES * H1);
    hipMemsetAsync(den1, 0, (size_t)N_NODES * H1 * 4, stream);
    hipMemsetAsync(agg1, 0, (size_t)N_NODES * HC1 * 4, stream);

    edge_max_kernel<<<edgeBlocks, 256, 0, stream>>>(srcI, dstI, als_1, ald_1, emax1, H1);
    edge_exp_kernel<<<edgeBlocks, 256, 0, stream>>>(srcI, dstI, als_1, ald_1, emax1,
                                                    eexp1, den1, H1);
    edge_scatter_kernel<<<E_TOT, 256, 0, stream>>>(srcI, dstI, h1, eexp1, den1, agg1, H1, C1);
    {
        dim3 grid((HC1 + 255) / 256, N_NODES);
        bias_relu_kernel<<<grid, 256, 0, stream>>>(agg1, b1, HC1);
    }

    // ---------------- Layer 2 ----------------
    {   // h2 = relu(agg1) @ W2   (M=20000, N=384, K=768)
        dim3 grid((tilesM + 7) / 8, C2 / 64);
        wmma_gemm_f32<<<grid, 256, 0, stream>>>(agg1, W2, h2, N_NODES, C2, HC1);
    }
    logits_kernel<<<(N_NODES + 255) / 256, 256, 0, stream>>>(h2, as2, ad2, als_2, ald_2,
                                                             N_NODES, 1, C2);
    fill_neginf<<<(N_NODES + 255) / 256, 256, 0, stream>>>(emax2, N_NODES);
    hipMemsetAsync(den2, 0, (size_t)N_NODES * 4, stream);
    hipMemsetAsync(out, 0, (size_t)N_NODES * C2 * 4, stream);

    edge_max_kernel<<<edgeBlocks, 256, 0, stream>>>(srcI, dstI, als_2, ald_2, emax2, 1);
    edge_exp_kernel<<<edgeBlocks, 256, 0, stream>>>(srcI, dstI, als_2, ald_2, emax2,
                                                    eexp2, den2, 1);
    edge_scatter_kernel<<<E_TOT, 256, 0, stream>>>(srcI, dstI, h2, eexp2, den2, out, 1, C2);
    {
        dim3 grid((C2 + 255) / 256, N_NODES);
        bias_relu_kernel<<<grid, 256, 0, stream>>>(out, b2, C2);
    }
}